// GraphAtt_5609227288945
// MI455X (gfx1250) — compile-verified
//
#include <hip/hip_runtime.h>
#include <hip/hip_bf16.h>
#include <math.h>

typedef __attribute__((ext_vector_type(2))) float v2f;
typedef __attribute__((ext_vector_type(8))) float v8f;

#define DIM 256
#define KSLAB 32
#define NSLABS (DIM / KSLAB)
#define KNBR 32
#define NEGBIG (-1000000.0f)

// Async global->LDS copy of 16 bytes (CDNA5 GLOBAL_LOAD_ASYNC_TO_LDS_B128,
// tracked by ASYNCcnt). lds_off = LDS byte offset (VGPR), gptr = 64-bit
// global address (VGPR pair).
__device__ __forceinline__ void async_copy_b128(unsigned int lds_off,
                                                const void* gptr) {
  asm volatile("global_load_async_to_lds_b128 %0, %1, off"
               :: "v"(lds_off), "v"(gptr)
               : "memory");
}

__device__ __forceinline__ void wait_asynccnt0() {
  asm volatile("s_wait_asynccnt 0x0" ::: "memory");
}

__device__ __forceinline__ unsigned int lds_off_of(const void* shared_ptr) {
  // For LDS-aperture flat addresses, addr[31:0] is the LDS byte offset.
  return (unsigned int)(unsigned long long)shared_ptr;
}

// ---------------------------------------------------------------------------
// Kernel 1: proj = leaky_relu(word_vec @ W1^T, 0.2)  via V_WMMA_F32_16X16X4_F32
// Block: 256 threads (8 waves). Tile: 64 rows x 256 cols.
// K staged in LDS with double-buffered ASYNC global->LDS copies.
// Wave w: m-tile = w&3 (16 rows), n-group = w>>2 (8 n-tiles of 16 cols).
// ---------------------------------------------------------------------------
__global__ __launch_bounds__(256) void proj_gemm_kernel(
    const float* __restrict__ wv,   // [N, 256]
    const float* __restrict__ W1,   // [256, 256]
    float* __restrict__ out,        // [N, 256]
    int nrows) {
  __shared__ __align__(16) float As[2][64 * KSLAB];    // 2 x 8 KB   [m][k]
  __shared__ __align__(16) float Bs[2][DIM * KSLAB];   // 2 x 32 KB  [col][k]

  const int tid  = threadIdx.x;
  const int wave = tid >> 5;
  const int lane = tid & 31;
  const int rowBase = blockIdx.x * 64;

  const int mt = wave & 3;   // which 16-row m-tile
  const int ng = wave >> 2;  // which group of 8 n-tiles

  // Issue async staging of one K-slab (32 wide) into LDS buffer `buf`.
  auto stage = [&](int buf, int kk) {
    // A tile: 64 rows x 32 k = 512 x b128; 2 per thread. Clamp OOB rows.
    for (int i = tid; i < 64 * (KSLAB / 4); i += 256) {
      int r = i >> 3;            // 8 x b128 per row
      int c = (i & 7) << 2;
      int row = rowBase + r;
      if (row >= nrows) row = nrows - 1;   // clamp (stores are guarded)
      async_copy_b128(lds_off_of(&As[buf][r * KSLAB + c]),
                      wv + (size_t)row * DIM + kk + c);
    }
    // B tile: 256 cols x 32 k of W1 = 2048 x b128; 8 per thread.
    for (int i = tid; i < DIM * (KSLAB / 4); i += 256) {
      int col = i >> 3;
      int c = (i & 7) << 2;
      async_copy_b128(lds_off_of(&Bs[buf][col * KSLAB + c]),
                      W1 + (size_t)col * DIM + kk + c);
    }
  };

  v8f acc[8];
#pragma unroll
  for (int i = 0; i < 8; ++i) acc[i] = (v8f)0.0f;

  // Prologue: stage slab 0.
  stage(0, 0);
  wait_asynccnt0();
  __syncthreads();

  // A fragment addressing per ISA 16x4 f32 layout:
  //   lanes 0-15:  M=lane,    VGPR0 K=k, VGPR1 K=k+1
  //   lanes 16-31: M=lane-16, VGPR0 K=k+2, VGPR1 K=k+3
  const int mrow  = mt * 16 + (lane & 15);
  const int khalf = (lane >> 4) * 2;

  for (int s = 0; s < NSLABS; ++s) {
    const int cur = s & 1;
    // Overlap: kick off the next slab's async copies before computing.
    // buf[1-cur] was last READ in iteration s-1, and the end-of-iteration
    // barrier below ordered those reads before these writes.
    if (s + 1 < NSLABS) stage(1 - cur, (s + 1) * KSLAB);

    const float* Ab = As[cur];
    const float* Bb = Bs[cur];
#pragma unroll
    for (int k4 = 0; k4 < KSLAB; k4 += 4) {
      v2f a;
      a.x = Ab[mrow * KSLAB + k4 + khalf];
      a.y = Ab[mrow * KSLAB + k4 + khalf + 1];
#pragma unroll
      for (int t = 0; t < 8; ++t) {
        const int col = (ng * 8 + t) * 16 + (lane & 15);
        v2f b;
        b.x = Bb[col * KSLAB + k4 + khalf];
        b.y = Bb[col * KSLAB + k4 + khalf + 1];
        acc[t] = __builtin_amdgcn_wmma_f32_16x16x4_f32(
            /*neg_a=*/false, a, /*neg_b=*/false, b,
            /*c_mod=*/(short)0, acc[t], /*reuse_a=*/false, /*reuse_b=*/false);
      }
    }

    if (s + 1 < NSLABS) wait_asynccnt0();   // our async writes landed
    __syncthreads();                        // publish LDS across waves
  }

  // C/D layout: VGPR v -> M = v + 8*(lane>=16), N = lane&15.
  const int rhalf = (lane >> 4) * 8;
#pragma unroll
  for (int t = 0; t < 8; ++t) {
    const int col = (ng * 8 + t) * 16 + (lane & 15);
#pragma unroll
    for (int v = 0; v < 8; ++v) {
      const int row = rowBase + mt * 16 + rhalf + v;
      if (row < nrows) {
        float x = acc[t][v];
        out[(size_t)row * DIM + col] = (x > 0.0f) ? x : 0.2f * x;
      }
    }
  }
}

// ---------------------------------------------------------------------------
// Kernel 2: attention weights + aggregation. One block (256 thr) per b.
// Reads proj from d_out (pre-overwrite values), writes agg rows to d_ws.
// ---------------------------------------------------------------------------
__global__ __launch_bounds__(256) void attention_kernel(
    const float* __restrict__ wv,     // [N, 256]
    const float* __restrict__ Waux,   // [1, 4]
    const float* __restrict__ aux,    // [B, K, 4]
    const int*   __restrict__ srcIdx, // [B]
    const int*   __restrict__ neighs, // [B, K]
    const int*   __restrict__ maskv,  // [B, K]
    const float* __restrict__ proj,   // [N, 256] (== d_out)
    float*       __restrict__ aggOut) // [B, 256] (== d_ws)
{
  __shared__ float sq[DIM];
  __shared__ int   snb[KNBR];
  __shared__ int   smask[KNBR];
  __shared__ float sscore[KNBR];
  __shared__ float saux[KNBR];
  __shared__ float sw[KNBR];

  const int b   = blockIdx.x;
  const int tid = threadIdx.x;

  if (tid < KNBR) {
    snb[tid]   = neighs[(size_t)b * KNBR + tid];
    smask[tid] = maskv[(size_t)b * KNBR + tid];
    const float* a4 = aux + ((size_t)b * KNBR + tid) * 4;
    float z = a4[0] * Waux[0] + a4[1] * Waux[1] + a4[2] * Waux[2] + a4[3] * Waux[3];
    saux[tid] = 1.0f / (1.0f + expf(-z));   // sigmoid
  }
  const int srow = srcIdx[b];
  sq[tid] = wv[(size_t)srow * DIM + tid];
  __syncthreads();

  // scores[kn] = 5 * dot(q, word_vec[neigh]); 8 threads per neighbor.
  const int kn  = tid >> 3;
  const int sub = tid & 7;
  {
    const float* krow = wv + (size_t)snb[kn] * DIM + sub * 32;
    const float* qrow = sq + sub * 32;
    float part = 0.0f;
#pragma unroll
    for (int i = 0; i < 32; i += 4) {
      float4 kv = *(const float4*)(krow + i);
      float4 qv = *(const float4*)(qrow + i);
      part += kv.x * qv.x + kv.y * qv.y + kv.z * qv.z + kv.w * qv.w;
    }
    part += __shfl_xor(part, 1, 32);
    part += __shfl_xor(part, 2, 32);
    part += __shfl_xor(part, 4, 32);
    if (sub == 0) sscore[kn] = part * 5.0f;
  }
  __syncthreads();

  // Dual masked softmax over K=32, done by wave 0 (one element per lane).
  if (tid < 32) {
    const bool m = (smask[tid] == 1);
    float s1 = m ? sscore[tid] : NEGBIG;
    float s2 = m ? saux[tid]   : NEGBIG;
    float m1 = s1, m2 = s2;
#pragma unroll
    for (int o = 16; o > 0; o >>= 1) {
      m1 = fmaxf(m1, __shfl_xor(m1, o, 32));
      m2 = fmaxf(m2, __shfl_xor(m2, o, 32));
    }
    float e1 = expf(s1 - m1), e2 = expf(s2 - m2);
    float t1 = e1, t2 = e2;
#pragma unroll
    for (int o = 16; o > 0; o >>= 1) {
      t1 += __shfl_xor(t1, o, 32);
      t2 += __shfl_xor(t2, o, 32);
    }
    sw[tid] = 0.5f * (e1 / t1 + e2 / t2);
  }
  __syncthreads();

  // agg[b, d] = sum_k w[k] * proj[neigh_k, d]; d = tid (coalesced, L2-resident).
  float accv = 0.0f;
#pragma unroll 4
  for (int k = 0; k < KNBR; ++k) {
    accv += sw[k] * proj[(size_t)snb[k] * DIM + tid];
  }
  aggOut[(size_t)b * DIM + tid] = accv;
}

// ---------------------------------------------------------------------------
extern "C" void kernel_launch(void* const* d_in, const int* in_sizes, int n_in,
                              void* d_out, int out_size, void* d_ws, size_t ws_size,
                              hipStream_t stream) {
  const float* wv     = (const float*)d_in[0];   // word_vec [N,256]
  const float* W1     = (const float*)d_in[1];   // [256,256]
  const float* Waux   = (const float*)d_in[2];   // [1,4]
  const float* aux    = (const float*)d_in[3];   // [B,K,4]
  const int*   srcIdx = (const int*)d_in[4];     // [B]
  const int*   neighs = (const int*)d_in[5];     // [B,K]
  const int*   maskv  = (const int*)d_in[6];     // [B,K]

  float* out = (float*)d_out;   // [N,256]
  float* agg = (float*)d_ws;    // [B,256] scratch

  const int N = in_sizes[0] / DIM;   // 100000
  const int B = in_sizes[4];         // 20000

  // 1) proj -> d_out (full [N,256])
  proj_gemm_kernel<<<(N + 63) / 64, 256, 0, stream>>>(wv, W1, out, N);

  // 2) attention: reads proj from d_out (pre-overwrite), agg -> d_ws
  attention_kernel<<<B, 256, 0, stream>>>(wv, Waux, aux, srcIdx, neighs, maskv,
                                          out, agg);

  // 3) overwrite rows [0, B) of d_out with agg (stream-ordered after kernel 2)
  hipMemcpyAsync(out, agg, (size_t)B * DIM * sizeof(float),
                 hipMemcpyDeviceToDevice, stream);
}